// Block_89807766159607
// MI455X (gfx1250) — compile-verified
//
#include <hip/hip_runtime.h>
#include <math.h>

// MI455X / gfx1250 implementation of the AFNO + MoE block.
// Strategy:
//  - FFT replaced by truncated-mode DFT GEMMs (only modes [:32,:32] matter).
//  - All GEMMs (DFT stages, 1x1 convs) use V_WMMA_F32_16X16X4_F32 (fp32 matrix path).
//  - Top-4 gating: zero-gate experts skipped per-batch (workgroup-uniform branch).
//  - GroupNorms fused into producer/consumer kernels; MoE convs share one LDS tile.
//  - MoE tile staging uses GLOBAL_LOAD_ASYNC_TO_LDS_B128 (ASYNCcnt path).

typedef float v2f __attribute__((ext_vector_type(2)));
typedef float v8f __attribute__((ext_vector_type(8)));

#define DEVINL __device__ __forceinline__

#if __has_builtin(__builtin_amdgcn_global_load_async_to_lds_b128)
#define HAVE_ASYNC_LDS 1
// Parameter types per hipcc diagnostic: (AS1 v4i*, AS3 v4i*, imm offset, imm cpol)
typedef int v4i_vs __attribute__((vector_size(4 * sizeof(int))));
typedef __attribute__((address_space(1))) v4i_vs* gas_b128;
typedef __attribute__((address_space(3))) v4i_vs* las_b128;
#endif

namespace {

constexpr int kB = 8, kC = 64, kH = 192, kW = 192;
constexpr int kHW = kH * kW;             // 36864
constexpr int kM = 32;                   // retained modes
constexpr int kG = 8;                    // groupnorm groups
constexpr int kCpG = kC / kG;            // 8
constexpr float kEps = 1e-5f;
constexpr int kGNS = 32;                 // slices for groupnorm partial reduction
constexpr int kGNchunk = kCpG * kHW / kGNS;   // 9216
constexpr int LDSTR = 132;               // padded LDS row stride (floats)
constexpr double kPI = 3.14159265358979323846;

// ---- workspace layout (floats) -------------------------------------------
constexpr size_t OFF_CF = 0;                                   // 32x192 fwd/inv cos
constexpr size_t OFF_SF = OFF_CF + (size_t)kM * kW;            // 32x192 sin
constexpr size_t OFF_CI = OFF_SF + (size_t)kM * kW;            // inverse col cos (c_l/192 folded)
constexpr size_t OFF_SI = OFF_CI + (size_t)kM * kW;            // inverse col sin
constexpr size_t OFF_GNPART = OFF_SI + (size_t)kM * kW;        // 8*8*32*2 = 4096
constexpr size_t OFF_STATS1 = OFF_GNPART + (size_t)kB * kG * kGNS * 2;  // 128
constexpr size_t OFF_STATS2 = OFF_STATS1 + (size_t)kB * kG * 2;
constexpr size_t OFF_GF     = OFF_STATS2 + (size_t)kB * kG * 2;         // 512
constexpr size_t OFF_GATE   = OFF_GF + (size_t)kB * kC;                 // 128
constexpr size_t IMGN = (size_t)kB * kC;                                // 512 images
constexpr size_t OFF_TRR = 32768;                      // (b,c,w=192,k=32)  (reused as PrR)
constexpr size_t OFF_TRI = OFF_TRR + IMGN * 192 * 32;  // (reused as PrI)
constexpr size_t OFF_MTR = OFF_TRI + IMGN * 192 * 32;  // (b,c,l,k) 32x32
constexpr size_t OFF_MTI = OFF_MTR + IMGN * 1024;
constexpr size_t OFF_M2R = OFF_MTI + IMGN * 1024;      // (b,c,k,l) 32x32 post-MLP
constexpr size_t OFF_M2I = OFF_M2R + IMGN * 1024;
constexpr size_t OFF_H2  = OFF_M2I + IMGN * 1024;      // (B,C,H,W) residual stream
constexpr size_t OFF_FEAT = OFF_H2 + (size_t)kB * kC * kHW;   // (B,C,H,W) feats

DEVINL v8f wmma4(v2f a, v2f b, v8f c) {
  // D = A(16x4, f32) * B(4x16, f32) + C(16x16, f32); f32 NEG bits must be 0.
  return __builtin_amdgcn_wmma_f32_16x16x4_f32(false, a, false, b, (short)0, c,
                                               false, false);
}

DEVINL float gelu_f(float x) {
  return 0.5f * x * (1.0f + erff(x * 0.70710678118654752f));
}

DEVINL void wait_async0() {
#if defined(HAVE_ASYNC_LDS)
#if __has_builtin(__builtin_amdgcn_s_wait_asynccnt)
  __builtin_amdgcn_s_wait_asynccnt(0);
#else
  asm volatile("s_wait_asynccnt 0" ::: "memory");
#endif
#endif
}

// ---- DFT factor tables ----------------------------------------------------
__global__ void k_init_tabs(float* cf, float* sf, float* ci, float* si) {
  int idx = blockIdx.x * 256 + threadIdx.x;
  if (idx >= kM * kW) return;
  int k = idx / kW, h = idx % kW;
  double ang = 2.0 * kPI * (double)k * (double)h / (double)kW;
  float c = (float)cos(ang), s = (float)sin(ang);
  cf[idx] = c;
  sf[idx] = s;
  double cl = (k == 0) ? 1.0 : 2.0;   // conj-symmetric column doubling
  ci[idx] = (float)(cl * cos(ang) / 192.0);  // 1/sqrt(H*W) ortho norm folded
  si[idx] = (float)(cl * sin(ang) / 192.0);
}

// ---- GroupNorm statistics -------------------------------------------------
__global__ void k_gn_partial(const float* __restrict__ src, float* __restrict__ part) {
  int bid = blockIdx.x;                 // B*G*kGNS blocks
  int slice = bid % kGNS, bg = bid / kGNS;
  size_t base = (size_t)bg * kCpG * kHW + (size_t)slice * kGNchunk;
  int tid = threadIdx.x;
  float s = 0.f, q = 0.f;
  for (int i = tid; i < kGNchunk; i += 256) {
    float v = src[base + i];
    s += v; q += v * v;
  }
  __shared__ float ls[256], lq[256];
  ls[tid] = s; lq[tid] = q;
  __syncthreads();
  for (int st = 128; st > 0; st >>= 1) {
    if (tid < st) { ls[tid] += ls[tid + st]; lq[tid] += lq[tid + st]; }
    __syncthreads();
  }
  if (tid == 0) { part[bid * 2] = ls[0]; part[bid * 2 + 1] = lq[0]; }
}

__global__ void k_gn_finalize(const float* __restrict__ part, float* __restrict__ stats) {
  int bg = threadIdx.x;
  if (bg >= kB * kG) return;
  float s = 0.f, q = 0.f;
  for (int i = 0; i < kGNS; i++) {
    s += part[(bg * kGNS + i) * 2];
    q += part[(bg * kGNS + i) * 2 + 1];
  }
  const float inv = 1.0f / (float)(kCpG * kHW);
  float mu = s * inv;
  float var = q * inv - mu * mu;
  stats[bg * 2] = mu;
  stats[bg * 2 + 1] = rsqrtf(var + kEps);
}

// ---- Forward DFT stage 1: T[k,w] = sum_h gn1(x[h,w]) e^{-2pi i hk/H} -----
// Output transposed: Tr/Ti stored as [w][k] per image for coalesced stage-2 B.
__global__ void k_fwd1(const float* __restrict__ x, const float* __restrict__ stats1,
                       const float* __restrict__ gnw, const float* __restrict__ gnb,
                       const float* __restrict__ cf, const float* __restrict__ sf,
                       float* __restrict__ TrR, float* __restrict__ TrI) {
  int img = blockIdx.x;
  int b = img / kC, c = img % kC;
  int bg = b * kG + c / kCpG;
  float mu = stats1[bg * 2], rs = stats1[bg * 2 + 1];
  float gw = gnw[c] * rs;
  float gb = gnb[c] - mu * gw;
  const float* X = x + (size_t)img * kHW;
  float* TR = TrR + (size_t)img * 192 * 32;
  float* TI = TrI + (size_t)img * 192 * 32;

  int lane = threadIdx.x & 31, wv = threadIdx.x >> 5;
  int l16 = lane & 15, hh = lane >> 4;
  int m0 = (wv & 1) * 16;                       // k-tile
  for (int t = 0; t < 3; t++) {
    int n0 = ((wv >> 1) + 4 * t) * 16;          // w-tile (12 total)
    v8f accR = {}, accI = {};
    for (int ks = 0; ks < 48; ks++) {
      int k0 = ks * 4;
      int ar = (m0 + l16) * kW + k0 + 2 * hh;
      v2f ac, as_;
      ac.x = cf[ar];  ac.y = cf[ar + 1];
      as_.x = sf[ar]; as_.y = sf[ar + 1];
      int br = (k0 + 2 * hh) * kW + n0 + l16;
      v2f bx;
      bx.x = X[br] * gw + gb;
      bx.y = X[br + kW] * gw + gb;
      accR = wmma4(ac, bx, accR);
      accI = wmma4(as_, bx, accI);
    }
    for (int r = 0; r < 8; r++) {
      int row = m0 + r + 8 * hh;   // k
      int col = n0 + l16;          // w
      TR[col * 32 + row] = accR[r];
      TI[col * 32 + row] = -accI[r];
    }
  }
}

// ---- Forward DFT stage 2: M[k,l] = (1/192) sum_w T[k,w] e^{-2pi i wl/W} --
// Output layout Mt[l][k] per image.
__global__ void k_fwd2(const float* __restrict__ cf, const float* __restrict__ sf,
                       const float* __restrict__ TrR, const float* __restrict__ TrI,
                       float* __restrict__ MtR, float* __restrict__ MtI) {
  int img = blockIdx.x;
  const float* TR = TrR + (size_t)img * 192 * 32;
  const float* TI = TrI + (size_t)img * 192 * 32;
  int lane = threadIdx.x & 31, wv = threadIdx.x >> 5;  // 4 waves
  int l16 = lane & 15, hh = lane >> 4;
  int m0 = (wv & 1) * 16;    // l-tile
  int n0 = (wv >> 1) * 16;   // k-tile
  v8f accR = {}, accI = {};
  for (int ks = 0; ks < 48; ks++) {
    int k0 = ks * 4;
    int ar = (m0 + l16) * kW + k0 + 2 * hh;
    v2f ac, as_;
    ac.x = cf[ar];  ac.y = cf[ar + 1];
    as_.x = sf[ar]; as_.y = sf[ar + 1];
    int br = (k0 + 2 * hh) * 32 + n0 + l16;
    v2f bR, bI;
    bR.x = TR[br]; bR.y = TR[br + 32];
    bI.x = TI[br]; bI.y = TI[br + 32];
    accR = wmma4(ac, bR, accR);
    accR = wmma4(as_, bI, accR);
    accI = wmma4(ac, bI, accI);
    accI = wmma4(as_, -bR, accI);
  }
  const float sc = 1.0f / 192.0f;
  size_t base = (size_t)img * 1024;
  for (int r = 0; r < 8; r++) {
    int row = m0 + r + 8 * hh;   // l
    int col = n0 + l16;          // k
    MtR[base + row * 32 + col] = accR[r] * sc;
    MtI[base + row * 32 + col] = accI[r] * sc;
  }
}

// ---- Block-diagonal complex 2-layer MLP on modes (tiny, VALU) -------------
// Reads Mt[l][k]; writes M2[k][l] (k-major for coalesced inverse-stage-1 B).
__global__ void k_mlp(const float* __restrict__ MtR, const float* __restrict__ MtI,
                      const float* __restrict__ w1, const float* __restrict__ b1,
                      const float* __restrict__ w2, const float* __restrict__ b2,
                      float* __restrict__ M2R, float* __restrict__ M2I) {
  __shared__ float W1[2048], W2[2048], B1[128], B2[128];
  int tid = threadIdx.x;
  for (int i = tid; i < 2048; i += 256) { W1[i] = w1[i]; W2[i] = w2[i]; }
  for (int i = tid; i < 128; i += 256) { B1[i] = b1[i]; B2[i] = b2[i]; }
  __syncthreads();

  int gidx = blockIdx.x * 256 + tid;     // 8 * 1024 threads total
  int b = gidx >> 10;
  int mode = gidx & 1023;                // mode = l*32 + k in Mt layout
  int l = mode >> 5, k = mode & 31;

  for (int nb = 0; nb < 4; nb++) {
    float xr[16], xi[16];
    for (int i = 0; i < 16; i++) {
      size_t ad = ((size_t)(b * kC + nb * 16 + i)) * 1024 + mode;
      xr[i] = MtR[ad]; xi[i] = MtI[ad];
    }
    float o1r[16], o1i[16];
    for (int o = 0; o < 16; o++) {
      float sr = B1[nb * 16 + o], si = B1[64 + nb * 16 + o];
      for (int i = 0; i < 16; i++) {
        float wr = W1[(nb * 16 + i) * 16 + o];
        float wi = W1[1024 + (nb * 16 + i) * 16 + o];
        sr += xr[i] * wr - xi[i] * wi;
        si += xi[i] * wr + xr[i] * wi;
      }
      o1r[o] = gelu_f(sr); o1i[o] = gelu_f(si);
    }
    for (int o = 0; o < 16; o++) {
      float sr = B2[nb * 16 + o], si = B2[64 + nb * 16 + o];
      for (int i = 0; i < 16; i++) {
        float wr = W2[(nb * 16 + i) * 16 + o];
        float wi = W2[1024 + (nb * 16 + i) * 16 + o];
        sr += o1r[i] * wr - o1i[i] * wi;
        si += o1i[i] * wr + o1r[i] * wi;
      }
      size_t ad = ((size_t)(b * kC + nb * 16 + o)) * 1024 + (size_t)k * 32 + l;
      M2R[ad] = sr; M2I[ad] = si;
    }
  }
}

// ---- Inverse stage 1: P[h,l] = sum_k M[k,l] e^{+2pi i hk/H} ---------------
// Pr = sum Mr*cos - Mi*sin ; Pi = sum Mr*sin + Mi*cos. Output Pr/Pi[h][l].
__global__ void k_inv1(const float* __restrict__ cf, const float* __restrict__ sf,
                       const float* __restrict__ M2R, const float* __restrict__ M2I,
                       float* __restrict__ PrR, float* __restrict__ PrI) {
  int img = blockIdx.x;
  const float* MR = M2R + (size_t)img * 1024;
  const float* MI = M2I + (size_t)img * 1024;
  float* PR = PrR + (size_t)img * 192 * 32;
  float* PI = PrI + (size_t)img * 192 * 32;
  int lane = threadIdx.x & 31, wv = threadIdx.x >> 5;
  int l16 = lane & 15, hh = lane >> 4;
  for (int t = 0; t < 3; t++) {
    int tt = wv + 8 * t;               // 24 tiles
    int m0 = (tt >> 1) * 16;           // h-tile
    int n0 = (tt & 1) * 16;            // l-tile
    v8f accR = {}, accI = {};
    for (int ks = 0; ks < 8; ks++) {
      int k0 = ks * 4;
      int ar = (k0 + 2 * hh) * kW + m0 + l16;   // A[M=h][K=k] = cf[k][h]
      v2f ac, as_;
      ac.x = cf[ar];  ac.y = cf[ar + kW];
      as_.x = sf[ar]; as_.y = sf[ar + kW];
      int br = (k0 + 2 * hh) * 32 + n0 + l16;   // B[K=k][N=l]
      v2f bR, bI;
      bR.x = MR[br]; bR.y = MR[br + 32];
      bI.x = MI[br]; bI.y = MI[br + 32];
      accR = wmma4(ac, bR, accR);
      accR = wmma4(as_, -bI, accR);
      accI = wmma4(as_, bR, accI);
      accI = wmma4(ac, bI, accI);
    }
    for (int r = 0; r < 8; r++) {
      int row = m0 + r + 8 * hh;  // h
      int col = n0 + l16;         // l
      PR[row * 32 + col] = accR[r];
      PI[row * 32 + col] = accI[r];
    }
  }
}

// ---- Inverse stage 2 + residual: h2 = irfft + gn1(x) + x ------------------
__global__ void k_inv2res(const float* __restrict__ ci, const float* __restrict__ si,
                          const float* __restrict__ PrR, const float* __restrict__ PrI,
                          const float* __restrict__ x, const float* __restrict__ stats1,
                          const float* __restrict__ gnw, const float* __restrict__ gnb,
                          float* __restrict__ h2) {
  int img = blockIdx.x;
  int b = img / kC, c = img % kC;
  int bg = b * kG + c / kCpG;
  float mu = stats1[bg * 2], rs = stats1[bg * 2 + 1];
  float gw = gnw[c] * rs;
  float gb = gnb[c] - mu * gw;
  const float* PR = PrR + (size_t)img * 192 * 32;
  const float* PI = PrI + (size_t)img * 192 * 32;
  const float* X = x + (size_t)img * kHW;
  float* O = h2 + (size_t)img * kHW;
  int lane = threadIdx.x & 31, wv = threadIdx.x >> 5;
  int l16 = lane & 15, hh = lane >> 4;
  for (int t = 0; t < 18; t++) {
    int tt = wv + 8 * t;               // 144 tiles
    int m0 = (tt / 12) * 16;           // h-tile
    int n0 = (tt % 12) * 16;           // w-tile
    v8f acc = {};
    for (int ks = 0; ks < 8; ks++) {
      int k0 = ks * 4;
      int ar = (m0 + l16) * 32 + k0 + 2 * hh;   // A[M=h][K=l] = Pr[h][l]
      v2f aR, aI;
      aR.x = PR[ar]; aR.y = PR[ar + 1];
      aI.x = PI[ar]; aI.y = PI[ar + 1];
      int br = (k0 + 2 * hh) * kW + n0 + l16;   // B[K=l][N=w]
      v2f bc, bs;
      bc.x = ci[br]; bc.y = ci[br + kW];
      bs.x = si[br]; bs.y = si[br + kW];
      acc = wmma4(aR, bc, acc);
      acc = wmma4(-aI, bs, acc);
    }
    for (int r = 0; r < 8; r++) {
      int row = m0 + r + 8 * hh;
      int col = n0 + l16;
      int ad = row * kW + col;
      float xv = X[ad];
      O[ad] = acc[r] + xv + (xv * gw + gb);     // irfft + gn1(x) residual + x residual
    }
  }
}

// ---- fe conv: feats = gelu(W * gn2(h2) + b), GN2 fused into LDS staging ---
__global__ void k_conv_fe(const float* __restrict__ h2, const float* __restrict__ stats2,
                          const float* __restrict__ gnw, const float* __restrict__ gnb,
                          const float* __restrict__ few, const float* __restrict__ feb,
                          float* __restrict__ feat) {
  int b = blockIdx.y;
  int p0 = blockIdx.x * 128;
  __shared__ float lds[kC * LDSTR];
  int tid = threadIdx.x;
  for (int i = tid; i < kC * 128; i += 256) {
    int c = i >> 7, n = i & 127;
    float v = h2[((size_t)b * kC + c) * kHW + p0 + n];
    int bg = b * kG + c / kCpG;
    v = (v - stats2[bg * 2]) * stats2[bg * 2 + 1] * gnw[c] + gnb[c];
    lds[c * LDSTR + n] = v;
  }
  __syncthreads();

  int lane = tid & 31, wv = tid >> 5;
  int l16 = lane & 15, hh = lane >> 4;
  int m0 = (wv & 3) * 16;
  int nh = wv >> 2;
  v2f A[16];
  for (int i = 0; i < 16; i++) {
    int ar = (m0 + l16) * kC + 4 * i + 2 * hh;
    A[i].x = few[ar]; A[i].y = few[ar + 1];
  }
  v8f bias;
  for (int r = 0; r < 8; r++) bias[r] = feb[m0 + r + 8 * hh];
  for (int t = 0; t < 4; t++) {
    int nl0 = (nh * 4 + t) * 16;
    v8f acc = bias;
    for (int ks = 0; ks < 16; ks++) {
      int br = (ks * 4 + 2 * hh) * LDSTR + nl0 + l16;
      v2f bx; bx.x = lds[br]; bx.y = lds[br + LDSTR];
      acc = wmma4(A[ks], bx, acc);
    }
    for (int r = 0; r < 8; r++) {
      int row = m0 + r + 8 * hh, col = nl0 + l16;
      feat[((size_t)b * kC + row) * kHW + p0 + col] = gelu_f(acc[r]);
    }
  }
}

// ---- Global average pool of feats ----------------------------------------
__global__ void k_featmean(const float* __restrict__ feat, float* __restrict__ gf) {
  int img = blockIdx.x;   // b*64 + c
  int tid = threadIdx.x;
  const float* p = feat + (size_t)img * kHW;
  float s = 0.f;
  for (int i = tid; i < kHW; i += 256) s += p[i];
  __shared__ float ls[256];
  ls[tid] = s;
  __syncthreads();
  for (int st = 128; st > 0; st >>= 1) {
    if (tid < st) ls[tid] += ls[tid + st];
    __syncthreads();
  }
  if (tid == 0) gf[img] = ls[0] / (float)kHW;
}

// ---- Gating MLP + top-4 softmax (tiny) ------------------------------------
__global__ void k_gate(const float* __restrict__ gf,
                       const float* __restrict__ g1w, const float* __restrict__ g1b,
                       const float* __restrict__ bn1g, const float* __restrict__ bn1b,
                       const float* __restrict__ bn1rm, const float* __restrict__ bn1rv,
                       const float* __restrict__ caw1, const float* __restrict__ cab1,
                       const float* __restrict__ caw2, const float* __restrict__ cab2,
                       const float* __restrict__ g2w, const float* __restrict__ g2b,
                       const float* __restrict__ bn2g, const float* __restrict__ bn2b,
                       const float* __restrict__ bn2rm, const float* __restrict__ bn2rv,
                       const float* __restrict__ g3w, const float* __restrict__ g3b,
                       float* __restrict__ gate) {
  int b = blockIdx.x, tid = threadIdx.x;   // 128 threads
  __shared__ float h1[128], t8[8], h2s[64], sc[16];
  {
    float s = g1b[tid];
    for (int c = 0; c < kC; c++) s += gf[b * kC + c] * g1w[tid * kC + c];
    s = (s - bn1rm[tid]) * rsqrtf(bn1rv[tid] + kEps) * bn1g[tid] + bn1b[tid];
    h1[tid] = gelu_f(s);
  }
  __syncthreads();
  if (tid < 8) {
    float s = cab1[tid];
    for (int i = 0; i < 128; i++) s += h1[i] * caw1[tid * 128 + i];
    t8[tid] = gelu_f(s);
  }
  __syncthreads();
  {
    float s = cab2[tid];
    for (int j = 0; j < 8; j++) s += t8[j] * caw2[tid * 8 + j];
    h1[tid] *= 1.0f / (1.0f + expf(-2.0f * s));
  }
  __syncthreads();
  if (tid < 64) {
    float s = g2b[tid];
    for (int i = 0; i < 128; i++) s += h1[i] * g2w[tid * 128 + i];
    s = (s - bn2rm[tid]) * rsqrtf(bn2rv[tid] + kEps) * bn2g[tid] + bn2b[tid];
    h2s[tid] = gelu_f(s);
  }
  __syncthreads();
  if (tid < 16) {
    float s = g3b[tid];
    for (int i = 0; i < 64; i++) s += h2s[i] * g3w[tid * 64 + i];
    sc[tid] = s;
  }
  __syncthreads();
  if (tid == 0) {
    float v[16];
    for (int i = 0; i < 16; i++) v[i] = sc[i];
    int idx[4]; float val[4];
    for (int t = 0; t < 4; t++) {
      int bi = 0; float bv = v[0];
      for (int i = 1; i < 16; i++) if (v[i] > bv) { bv = v[i]; bi = i; }
      idx[t] = bi; val[t] = bv; v[bi] = -1e30f;
    }
    float mx = val[0], e[4], se = 0.f;
    for (int t = 0; t < 4; t++) { e[t] = expf((val[t] - mx) * 0.5f); se += e[t]; }
    float g[16];
    for (int i = 0; i < 16; i++) g[i] = 0.f;
    for (int t = 0; t < 4; t++) g[idx[t]] += e[t] / se;
    for (int i = 0; i < 16; i++) gate[b * 16 + i] = g[i];
  }
}

// ---- Fused MoE: out = sum_s 0.5*gelu(conv_s) + sum_e gate*gelu(conv_e) + h2
__global__ void k_moe(const float* __restrict__ feat, const float* __restrict__ h2,
                      const float* __restrict__ sharedw, const float* __restrict__ sharedb,
                      const float* __restrict__ expw, const float* __restrict__ expb,
                      const float* __restrict__ gate, float* __restrict__ out) {
  int b = blockIdx.y;
  int p0 = blockIdx.x * 128;
  __shared__ float lds[kC * LDSTR];
  int tid = threadIdx.x;
#if defined(HAVE_ASYNC_LDS)
  // CDNA5 async DMA path: global -> LDS without a VGPR round trip (ASYNCcnt).
  // 64 rows x 128 floats; each thread issues one b128 (4-float) transfer per
  // iteration. 16B alignment holds: rows start at multiples of 128 floats in
  // global and LDSTR=132 floats (528B) in LDS; in-row offsets are multiples of 16B.
  for (int i = tid * 4; i < kC * 128; i += 256 * 4) {
    int c = i >> 7, n = i & 127;
    __builtin_amdgcn_global_load_async_to_lds_b128(
        (gas_b128)(feat + ((size_t)b * kC + c) * kHW + p0 + n),
        (las_b128)&lds[c * LDSTR + n], 0, 0);
  }
  wait_async0();
#else
  for (int i = tid; i < kC * 128; i += 256) {
    int c = i >> 7, n = i & 127;
    lds[c * LDSTR + n] = feat[((size_t)b * kC + c) * kHW + p0 + n];
  }
#endif
  __syncthreads();

  int lane = tid & 31, wv = tid >> 5;
  int l16 = lane & 15, hh = lane >> 4;
  int m0 = (wv & 3) * 16;
  int nh = wv >> 2;
  v8f outa[4];
  for (int t = 0; t < 4; t++) outa[t] = (v8f){};

  for (int cv = 0; cv < 18; cv++) {
    const float* Wp; const float* Bp; float scale;
    if (cv < 2) {
      Wp = sharedw + cv * kC * kC; Bp = sharedb + cv * kC; scale = 0.5f;
    } else {
      int e = cv - 2;
      scale = gate[b * 16 + e];
      if (scale == 0.0f) continue;   // top-4 gating: skip inactive experts
      Wp = expw + (size_t)e * kC * kC; Bp = expb + e * kC;
    }
    v2f A[16];
    for (int i = 0; i < 16; i++) {
      int ar = (m0 + l16) * kC + 4 * i + 2 * hh;
      A[i].x = Wp[ar]; A[i].y = Wp[ar + 1];
    }
    v8f bias;
    for (int r = 0; r < 8; r++) bias[r] = Bp[m0 + r + 8 * hh];
    for (int t = 0; t < 4; t++) {
      int nl0 = (nh * 4 + t) * 16;
      v8f acc = bias;
      for (int ks = 0; ks < 16; ks++) {
        int br = (ks * 4 + 2 * hh) * LDSTR + nl0 + l16;
        v2f bx; bx.x = lds[br]; bx.y = lds[br + LDSTR];
        acc = wmma4(A[ks], bx, acc);
      }
      for (int r = 0; r < 8; r++) outa[t][r] += scale * gelu_f(acc[r]);
    }
  }
  for (int t = 0; t < 4; t++) {
    int nl0 = (nh * 4 + t) * 16;
    for (int r = 0; r < 8; r++) {
      int row = m0 + r + 8 * hh, col = nl0 + l16;
      size_t ad = ((size_t)b * kC + row) * kHW + p0 + col;
      out[ad] = outa[t][r] + h2[ad];
    }
  }
}

}  // namespace

extern "C" void kernel_launch(void* const* d_in, const int* in_sizes, int n_in,
                              void* d_out, int out_size, void* d_ws, size_t ws_size,
                              hipStream_t stream) {
  (void)in_sizes; (void)n_in; (void)out_size; (void)ws_size;
  const float* x       = (const float*)d_in[0];
  const float* gn1_w   = (const float*)d_in[1];
  const float* gn1_b   = (const float*)d_in[2];
  const float* afno_w1 = (const float*)d_in[3];
  const float* afno_b1 = (const float*)d_in[4];
  const float* afno_w2 = (const float*)d_in[5];
  const float* afno_b2 = (const float*)d_in[6];
  const float* gn2_w   = (const float*)d_in[7];
  const float* gn2_b   = (const float*)d_in[8];
  const float* fe_w    = (const float*)d_in[9];
  const float* fe_b    = (const float*)d_in[10];
  const float* g1_w    = (const float*)d_in[11];
  const float* g1_b    = (const float*)d_in[12];
  const float* bn1_g   = (const float*)d_in[13];
  const float* bn1_b   = (const float*)d_in[14];
  const float* bn1_rm  = (const float*)d_in[15];
  const float* bn1_rv  = (const float*)d_in[16];
  const float* ca_w1   = (const float*)d_in[17];
  const float* ca_b1   = (const float*)d_in[18];
  const float* ca_w2   = (const float*)d_in[19];
  const float* ca_b2   = (const float*)d_in[20];
  const float* g2_w    = (const float*)d_in[21];
  const float* g2_b    = (const float*)d_in[22];
  const float* bn2_g   = (const float*)d_in[23];
  const float* bn2_b   = (const float*)d_in[24];
  const float* bn2_rm  = (const float*)d_in[25];
  const float* bn2_rv  = (const float*)d_in[26];
  const float* g3_w    = (const float*)d_in[27];
  const float* g3_b    = (const float*)d_in[28];
  const float* shared_w= (const float*)d_in[29];
  const float* shared_b= (const float*)d_in[30];
  const float* expert_w= (const float*)d_in[31];
  const float* expert_b= (const float*)d_in[32];

  float* ws   = (float*)d_ws;
  float* cf   = ws + OFF_CF;
  float* sf   = ws + OFF_SF;
  float* ci   = ws + OFF_CI;
  float* si   = ws + OFF_SI;
  float* part = ws + OFF_GNPART;
  float* st1  = ws + OFF_STATS1;
  float* st2  = ws + OFF_STATS2;
  float* gf   = ws + OFF_GF;
  float* gate = ws + OFF_GATE;
  float* TrR  = ws + OFF_TRR;   // stage-1 fwd output; reused as PrR after fwd2
  float* TrI  = ws + OFF_TRI;   // reused as PrI
  float* MtR  = ws + OFF_MTR;
  float* MtI  = ws + OFF_MTI;
  float* M2R  = ws + OFF_M2R;
  float* M2I  = ws + OFF_M2I;
  float* h2   = ws + OFF_H2;
  float* feat = ws + OFF_FEAT;
  float* outp = (float*)d_out;

  const int NIMG = kB * kC;  // 512

  k_init_tabs<<<24, 256, 0, stream>>>(cf, sf, ci, si);
  k_gn_partial<<<kB * kG * kGNS, 256, 0, stream>>>(x, part);
  k_gn_finalize<<<1, 64, 0, stream>>>(part, st1);
  k_fwd1<<<NIMG, 256, 0, stream>>>(x, st1, gn1_w, gn1_b, cf, sf, TrR, TrI);
  k_fwd2<<<NIMG, 128, 0, stream>>>(cf, sf, TrR, TrI, MtR, MtI);
  k_mlp<<<32, 256, 0, stream>>>(MtR, MtI, afno_w1, afno_b1, afno_w2, afno_b2, M2R, M2I);
  k_inv1<<<NIMG, 256, 0, stream>>>(cf, sf, M2R, M2I, TrR, TrI);  // TrR/TrI reused as Pr
  k_inv2res<<<NIMG, 256, 0, stream>>>(ci, si, TrR, TrI, x, st1, gn1_w, gn1_b, h2);
  k_gn_partial<<<kB * kG * kGNS, 256, 0, stream>>>(h2, part);
  k_gn_finalize<<<1, 64, 0, stream>>>(part, st2);
  k_conv_fe<<<dim3(kHW / 128, kB), 256, 0, stream>>>(h2, st2, gn2_w, gn2_b, fe_w, fe_b, feat);
  k_featmean<<<NIMG, 256, 0, stream>>>(feat, gf);
  k_gate<<<kB, 128, 0, stream>>>(gf, g1_w, g1_b, bn1_g, bn1_b, bn1_rm, bn1_rv,
                                 ca_w1, ca_b1, ca_w2, ca_b2, g2_w, g2_b,
                                 bn2_g, bn2_b, bn2_rm, bn2_rv, g3_w, g3_b, gate);
  k_moe<<<dim3(kHW / 128, kB), 256, 0, stream>>>(feat, h2, shared_w, shared_b,
                                                 expert_w, expert_b, gate, outp);
}